// CompGCN_59871844106326
// MI455X (gfx1250) — compile-verified
//
#include <hip/hip_runtime.h>
#include <math.h>

// ---------------- problem constants (from reference) ----------------
#define N_NODES  50000
#define N_EDGES  250000
#define DIM      256
#define N_REL2   400
#define LN_EPS   1e-5f
#define PI_F     3.14159265358979323846f
#define E_TOTAL  (N_EDGES + N_NODES)   // edges + self loops = 300000

typedef float v2f __attribute__((ext_vector_type(2)));
typedef float v8f __attribute__((ext_vector_type(8)));

// ---------------- small helpers ----------------
__device__ __forceinline__ int brev8(int i) {
  return (int)(__brev((unsigned)i) >> 24);
}

__device__ __forceinline__ void softmax3(const float* cw, float& w0, float& w1, float& w2) {
  float c0 = cw[0], c1 = cw[1], c2 = cw[2];
  float m = fmaxf(c0, fmaxf(c1, c2));
  float e0 = expf(c0 - m), e1 = expf(c1 - m), e2 = expf(c2 - m);
  float inv = 1.0f / (e0 + e1 + e2);
  w0 = e0 * inv; w1 = e1 * inv; w2 = e2 * inv;
}

// LDS byte offset (LDS_BASE-relative) of a __shared__ object: low 32 bits of
// the generic pointer (aperture lives in the high half on gfx1250).
__device__ __forceinline__ unsigned lds_off(const void* p) {
  return (unsigned)(unsigned long long)(uintptr_t)p;
}

// async copy of 16B global -> LDS, tracked by ASYNCcnt
__device__ __forceinline__ void async_copy16(unsigned ldsByteOff, const void* gptr) {
  asm volatile("global_load_async_to_lds_b128 %0, %1, off"
               :: "v"(ldsByteOff), "v"(gptr) : "memory");
}

__device__ __forceinline__ void wait_async0() {
  asm volatile("s_wait_asynccnt 0x0" ::: "memory");
}

// ---------------- h = emb[x] gather (float4, coalesced) ----------------
__global__ __launch_bounds__(256) void gather_kernel(const int* __restrict__ x,
                                                     const float* __restrict__ emb,
                                                     float* __restrict__ h) {
  int idx = blockIdx.x * 256 + threadIdx.x;        // float4 index, N*64 total
  if (idx >= N_NODES * 64) return;
  int i  = idx >> 6;
  int k4 = idx & 63;
  const float4* s = (const float4*)(emb + (size_t)x[i] * DIM);
  ((float4*)(h + (size_t)i * DIM))[k4] = s[k4];
}

// ---------------- zero fill (float4) ----------------
__global__ __launch_bounds__(256) void zero_kernel(float4* __restrict__ p, int n4) {
  int idx = blockIdx.x * 256 + threadIdx.x;
  if (idx < n4) p[idx] = make_float4(0.f, 0.f, 0.f, 0.f);
}

// ---------------- forward FFT-256 of real rows (LDS radix-2 DIT) ----------------
// out[row][k] = sum_n in[row][n] e^{-2pi i nk/256};   conjOut=1 stores conj(X).
__global__ __launch_bounds__(128) void fft_rows_kernel(const float* __restrict__ in,
                                                       float2* __restrict__ out,
                                                       int conjOut) {
  __shared__ float2 s[256];
  __shared__ float2 tw[128];
  int tid = threadIdx.x;           // 0..127
  int row = blockIdx.x;

  float ang = -2.0f * PI_F * (float)tid / 256.0f;  // forward twiddles
  tw[tid] = make_float2(cosf(ang), sinf(ang));

  const float* p = in + (size_t)row * DIM;
  int i0 = tid, i1 = tid + 128;
  s[brev8(i0)] = make_float2(p[i0], 0.f);
  s[brev8(i1)] = make_float2(p[i1], 0.f);
  __syncthreads();

  int sh = 7;
  for (int len = 2; len <= 256; len <<= 1, sh--) {
    int half = len >> 1;
    int pos  = tid & (half - 1);
    int a    = ((tid ^ pos) << 1) | pos;   // grp*len + pos
    int b    = a + half;
    float2 w = tw[pos << sh];
    float2 u = s[a], v = s[b];
    float2 vw = make_float2(v.x * w.x - v.y * w.y, v.x * w.y + v.y * w.x);
    s[a] = make_float2(u.x + vw.x, u.y + vw.y);
    s[b] = make_float2(u.x - vw.x, u.y - vw.y);
    __syncthreads();
  }

  float sgn = conjOut ? -1.0f : 1.0f;
  float2* q = out + (size_t)row * DIM;
  q[i0] = make_float2(s[i0].x, sgn * s[i0].y);
  q[i1] = make_float2(s[i1].x, sgn * s[i1].y);
}

// ---------------- edge kernel: one wave32 per edge ----------------
// freq part:  aggfreq[dst] += FFT(h[src]) * conj(FFT(r[et]))   (rfreq pre-conjugated)
// time part:  agg_ma[dst]  += w0*(h*r) + w1*(h+r)
__global__ __launch_bounds__(256) void edge_kernel(const int* __restrict__ src,
                                                   const int* __restrict__ dst,
                                                   const int* __restrict__ et,
                                                   const float* __restrict__ h,
                                                   const float* __restrict__ rel,
                                                   const float2* __restrict__ hfreq,
                                                   const float2* __restrict__ rfreq,
                                                   float* __restrict__ aggfreq,
                                                   float* __restrict__ agg_ma,
                                                   const float* __restrict__ cw) {
  int wave = threadIdx.x >> 5;
  int lane = threadIdx.x & 31;
  int e = blockIdx.x * 8 + wave;
  if (e >= E_TOTAL) return;

  int s, d, t;
  if (e < N_EDGES) { s = src[e]; d = dst[e]; t = et[e]; }
  else             { s = e - N_EDGES; d = s; t = 0; }     // self loop, fill_value=0

  float w0, w1, w2;
  softmax3(cw, w0, w1, w2);
  (void)w2;

  const float2* hf = hfreq + (size_t)s * DIM;
  const float2* rf = rfreq + (size_t)t * DIM;
  const float*  hv = h     + (size_t)s * DIM;
  const float*  rv = rel   + (size_t)t * DIM;
  float* af = aggfreq + (size_t)d * (2 * DIM);
  float* am = agg_ma  + (size_t)d * DIM;

#pragma unroll
  for (int j = 0; j < 8; j++) {
    int k = lane + 32 * j;                 // coalesced across the wave
    float2 a = hf[k], b = rf[k];
    float pre = a.x * b.x - a.y * b.y;
    float pim = a.x * b.y + a.y * b.x;
    atomicAdd(&af[2 * k],     pre);
    atomicAdd(&af[2 * k + 1], pim);
    float xv = hv[k], rr = rv[k];
    atomicAdd(&am[k], w0 * xv * rr + w1 * (xv + rr));
  }
}

// ---------------- inverse FFT of aggfreq rows, combine into agg_ma ----------------
// agg_ma[row] += w2 * Re( IFFT(aggfreq[row]) )
__global__ __launch_bounds__(128) void ifft_combine_kernel(const float2* __restrict__ aggf,
                                                           float* __restrict__ agg_ma,
                                                           const float* __restrict__ cw) {
  __shared__ float2 s[256];
  __shared__ float2 tw[128];
  int tid = threadIdx.x;
  int row = blockIdx.x;

  float ang = 2.0f * PI_F * (float)tid / 256.0f;   // inverse twiddles
  tw[tid] = make_float2(cosf(ang), sinf(ang));

  const float2* p = aggf + (size_t)row * DIM;
  int i0 = tid, i1 = tid + 128;
  s[brev8(i0)] = p[i0];
  s[brev8(i1)] = p[i1];
  __syncthreads();

  int sh = 7;
  for (int len = 2; len <= 256; len <<= 1, sh--) {
    int half = len >> 1;
    int pos  = tid & (half - 1);
    int a    = ((tid ^ pos) << 1) | pos;
    int b    = a + half;
    float2 w = tw[pos << sh];
    float2 u = s[a], v = s[b];
    float2 vw = make_float2(v.x * w.x - v.y * w.y, v.x * w.y + v.y * w.x);
    s[a] = make_float2(u.x + vw.x, u.y + vw.y);
    s[b] = make_float2(u.x - vw.x, u.y - vw.y);
    __syncthreads();
  }

  float w0, w1, w2;
  softmax3(cw, w0, w1, w2);
  (void)w0; (void)w1;
  const float scale = w2 * (1.0f / 256.0f);
  float* am = agg_ma + (size_t)row * DIM;
  am[i0] += scale * s[i0].x;
  am[i1] += scale * s[i1].x;
}

// ---------------- WMMA f32 GEMM + fused bias/residual/LayerNorm/ReLU ----------------
// out[i][o] = relu(LN( A[i][:] . W[o][:] + bias[o] + resid[i][o] ))   (doLN=1)
// out[i][o] = A[i][:] . W[o][:] + bias[o]                             (doLN=0)
// Block: 256 threads (8 waves), one 16-row x 256-col output strip per block.
// W is staged into LDS with GLOBAL_LOAD_ASYNC_TO_LDS_B128 (ASYNCcnt-tracked),
// avoiding the VGPR round-trip; each wave waits its own ASYNCcnt before the
// workgroup barrier.
#define SA_STRIDE 260     // 16 rows of A, padded: bank-conflict-free fragment reads
#define SB_STRIDE 17      // 256 rows of W, K-chunk of 16, padded

__global__ __launch_bounds__(256) void gemm_ln_kernel(const float* __restrict__ A,
                                                      const float* __restrict__ W,
                                                      const float* __restrict__ bias,
                                                      const float* __restrict__ resid,
                                                      const float* __restrict__ gamma,
                                                      const float* __restrict__ beta,
                                                      float* __restrict__ out,
                                                      int doLN) {
  __shared__ float sA[16 * SA_STRIDE];     // A tile (later reused as output tile)
  __shared__ float sB[256 * SB_STRIDE];    // W K-chunk (all 256 output cols x 16 k)
  __shared__ float sRed[2 * 256];          // LN partial sums / sumsq
  __shared__ float sStat[32];              // mu, rstd per row

  int tid  = threadIdx.x;
  int lane = tid & 31;
  int wave = tid >> 5;
  int l2   = lane & 15;
  int hi   = lane >> 4;
  int cb   = wave * 32;                    // this wave's 32 output columns
  int i0   = blockIdx.x * 16;              // row block

  // ---- async-load A tile: 16x256 floats (256 x 16B per q-step, coalesced) ----
  {
    const float4* Ap = (const float4*)(A + (size_t)i0 * DIM);
#pragma unroll
    for (int q = 0; q < 4; q++) {
      int f4 = tid + q * 256;              // 0..1023
      int r  = f4 >> 6;
      int c4 = f4 & 63;
      async_copy16(lds_off(&sA[r * SA_STRIDE + c4 * 4]), (const void*)(Ap + f4));
    }
  }

  v8f acc0 = {0.f, 0.f, 0.f, 0.f, 0.f, 0.f, 0.f, 0.f};
  v8f acc1 = {0.f, 0.f, 0.f, 0.f, 0.f, 0.f, 0.f, 0.f};

  for (int kb0 = 0; kb0 < DIM; kb0 += 16) {
    __syncthreads();                       // previous chunk fully consumed
    // stage W[:, kb0:kb0+16] into LDS (4096 floats) via async copies
#pragma unroll
    for (int q = 0; q < 4; q++) {
      int g  = tid + q * 256;              // 0..1023
      int o  = g >> 2;
      int c4 = g & 3;
      async_copy16(lds_off(&sB[o * SB_STRIDE + c4 * 4]),
                   (const void*)(W + (size_t)o * DIM + kb0 + c4 * 4));
    }
    if (kb0 + 16 < DIM)
      __builtin_prefetch(W + (size_t)(tid) * DIM + kb0 + 16, 0, 1);
    wait_async0();                         // this wave's copies landed in LDS
    __syncthreads();                       // all waves' copies visible

#pragma unroll
    for (int kk = 0; kk < 16; kk += 4) {
      // A fragment (16x4): lane<16 -> K pair {kk,kk+1}; lane>=16 -> {kk+2,kk+3}
      int kA = kb0 + kk + 2 * hi;
      v2f a;
      a.x = sA[l2 * SA_STRIDE + kA];
      a.y = sA[l2 * SA_STRIDE + kA + 1];
      int kB = kk + 2 * hi;
      // B fragment (4x16): B[k][n] = W[n][k]
      v2f b0, b1;
      b0.x = sB[(cb + l2) * SB_STRIDE + kB];
      b0.y = sB[(cb + l2) * SB_STRIDE + kB + 1];
      b1.x = sB[(cb + 16 + l2) * SB_STRIDE + kB];
      b1.y = sB[(cb + 16 + l2) * SB_STRIDE + kB + 1];
      acc0 = __builtin_amdgcn_wmma_f32_16x16x4_f32(false, a, false, b0,
                                                   (short)0, acc0, false, false);
      acc1 = __builtin_amdgcn_wmma_f32_16x16x4_f32(false, a, false, b1,
                                                   (short)0, acc1, false, false);
    }
  }

  __syncthreads();
  // spill accumulators to LDS (reuse sA as the 16x256 output tile)
#pragma unroll
  for (int v = 0; v < 8; v++) {
    int m = v + 8 * hi;                       // C/D layout: VGPR v holds M=v / M=v+8
    sA[m * SA_STRIDE + cb + l2]      = acc0[v];
    sA[m * SA_STRIDE + cb + 16 + l2] = acc1[v];
  }
  __syncthreads();

  // ---- epilogue: bias (+ residual + LayerNorm + ReLU) ----
  int r   = tid & 15;                         // row within tile
  int seg = tid >> 4;                         // 16-col segment
  float vals[16];
  float psum = 0.f, psq = 0.f;
#pragma unroll
  for (int j = 0; j < 16; j++) {
    int n = seg * 16 + j;
    float v = sA[r * SA_STRIDE + n] + bias[n];
    if (resid) v += resid[(size_t)(i0 + r) * DIM + n];
    vals[j] = v;
    psum += v;
    psq  += v * v;
  }

  if (doLN) {
    sRed[r * 16 + seg]       = psum;
    sRed[256 + r * 16 + seg] = psq;
    __syncthreads();
    if (tid < 16) {
      float sum = 0.f, sq = 0.f;
#pragma unroll
      for (int j = 0; j < 16; j++) {
        sum += sRed[tid * 16 + j];
        sq  += sRed[256 + tid * 16 + j];
      }
      float mu  = sum * (1.0f / 256.0f);
      float var = sq * (1.0f / 256.0f) - mu * mu;
      sStat[tid]      = mu;
      sStat[16 + tid] = 1.0f / sqrtf(var + LN_EPS);
    }
    __syncthreads();
    float mu = sStat[r], rstd = sStat[16 + r];
#pragma unroll
    for (int j = 0; j < 16; j++) {
      int n = seg * 16 + j;
      float z = (vals[j] - mu) * rstd * gamma[n] + beta[n];
      out[(size_t)(i0 + r) * DIM + n] = fmaxf(z, 0.f);
    }
  } else {
#pragma unroll
    for (int j = 0; j < 16; j++) {
      int n = seg * 16 + j;
      out[(size_t)(i0 + r) * DIM + n] = vals[j];
    }
  }
}

// ---------------- host-side launcher ----------------
extern "C" void kernel_launch(void* const* d_in, const int* in_sizes, int n_in,
                              void* d_out, int out_size, void* d_ws, size_t ws_size,
                              hipStream_t stream) {
  (void)in_sizes; (void)n_in; (void)out_size;

  const int*   x     = (const int*)  d_in[0];
  const int*   eidx  = (const int*)  d_in[1];      // [2][E]
  const int*   etype = (const int*)  d_in[2];
  const float* emb   = (const float*)d_in[3];
  const float* rel1  = (const float*)d_in[4];
  // d_in[5]=Wr1, d_in[6]=br1: layer-1 rels output is discarded by the reference.
  const float* Wn1   = (const float*)d_in[7];
  const float* bn1   = (const float*)d_in[8];
  const float* cw1   = (const float*)d_in[9];
  const float* g1    = (const float*)d_in[10];
  const float* b1    = (const float*)d_in[11];
  const float* rel2  = (const float*)d_in[12];
  const float* Wr2   = (const float*)d_in[13];
  const float* br2   = (const float*)d_in[14];
  const float* Wn2   = (const float*)d_in[15];
  const float* bn2   = (const float*)d_in[16];
  const float* cw2   = (const float*)d_in[17];
  const float* g2    = (const float*)d_in[18];
  const float* b2    = (const float*)d_in[19];

  const int* src = eidx;
  const int* dst = eidx + N_EDGES;

  // workspace layout (floats): h | hfreq | aggfreq | agg_ma | rfreq  (~308 MB)
  float*  ws      = (float*)d_ws;
  float*  h       = ws;                           // 12,800,000
  float2* hfreq   = (float2*)(ws + 12800000);     // 25,600,000 floats
  float*  aggfreq = ws + 38400000;                // 25,600,000
  float*  agg_ma  = ws + 64000000;                // 12,800,000
  float2* rfreq   = (float2*)(ws + 76800000);     //    204,800 floats
  if (ws_size < (size_t)77004800 * sizeof(float)) return;  // insufficient scratch

  float* outH = (float*)d_out;
  float* outR = (float*)d_out + (size_t)N_NODES * DIM;

  // h = emb[x]
  gather_kernel<<<dim3(12500), dim3(256), 0, stream>>>(x, emb, h);

  for (int layer = 0; layer < 2; layer++) {
    const float* rel = layer ? rel2 : rel1;
    const float* Wn  = layer ? Wn2  : Wn1;
    const float* bn  = layer ? bn2  : bn1;
    const float* cw  = layer ? cw2  : cw1;
    const float* g   = layer ? g2   : g1;
    const float* bb  = layer ? b2   : b1;
    float* gout      = layer ? outH : h;          // layer1 overwrites h in-place

    // zero aggfreq + agg_ma (contiguous 38.4M floats = 9.6M float4)
    zero_kernel<<<dim3(37500), dim3(256), 0, stream>>>((float4*)aggfreq, 9600000);
    // per-row FFTs
    fft_rows_kernel<<<dim3(N_NODES), dim3(128), 0, stream>>>(h, hfreq, 0);
    fft_rows_kernel<<<dim3(N_REL2), dim3(128), 0, stream>>>(rel, rfreq, 1);  // conj
    // edges + self-loops: scatter in freq + time domains
    edge_kernel<<<dim3(E_TOTAL / 8), dim3(256), 0, stream>>>(
        src, dst, etype, h, rel, hfreq, rfreq, aggfreq, agg_ma, cw);
    // IFFT + weighted combine -> agg_ma holds full aggregation
    ifft_combine_kernel<<<dim3(N_NODES), dim3(128), 0, stream>>>(
        (const float2*)aggfreq, agg_ma, cw);
    // out = relu(LN(agg @ Wn^T + bn + h))
    gemm_ln_kernel<<<dim3(N_NODES / 16), dim3(256), 0, stream>>>(
        agg_ma, Wn, bn, h, g, bb, gout, 1);
  }

  // rels = rel_emb2 @ Wr2^T + br2
  gemm_ln_kernel<<<dim3(N_REL2 / 16), dim3(256), 0, stream>>>(
      rel2, Wr2, br2, nullptr, nullptr, nullptr, outR, 0);
}